// Net_21397527069336
// MI455X (gfx1250) — compile-verified
//
#include <hip/hip_runtime.h>

// ---------------------------------------------------------------------------
// 2-layer GCN, reordered so aggregation happens in the *narrow* feature space
// (4-dim before W1, 2-dim after W2) -- mathematically identical since GCN
// aggregation is linear. The only dense GEMM (agg4 @ W1, K=4) is done with
// V_WMMA_F32_16X16X4_F32 and fused with bias+ReLU+the 128->2 projection so
// the 100000x128 hidden matrix never touches memory.
// ---------------------------------------------------------------------------

typedef __attribute__((ext_vector_type(2))) float v2f;
typedef __attribute__((ext_vector_type(8))) float v8f;

// ---------------- edge/node utility kernels --------------------------------

__global__ void k_zero(float* p, int n) {
    int i = blockIdx.x * blockDim.x + threadIdx.x;
    if (i < n) p[i] = 0.0f;
}

__global__ void k_degree(const long long* __restrict__ ei, float* __restrict__ deg, int E) {
    int e = blockIdx.x * blockDim.x + threadIdx.x;
    if (e >= E) return;
    int d = (int)ei[E + e];             // dst row of edge_index
    atomicAdd(&deg[d], 1.0f);
}

// dinv = rsqrt(deg+1) written in place over deg; agg4 initialized with the
// self-loop term x[v]*dinv^2 (non-atomic init, precedes edge atomics).
__global__ void k_dinv_selfloop(float* __restrict__ deg_dinv,
                                const float* __restrict__ x,
                                float* __restrict__ agg4, int n) {
    int v = blockIdx.x * blockDim.x + threadIdx.x;
    if (v >= n) return;
    float di = rsqrtf(deg_dinv[v] + 1.0f);
    deg_dinv[v] = di;
    float s = di * di;
    agg4[v * 4 + 0] = x[v * 4 + 0] * s;
    agg4[v * 4 + 1] = x[v * 4 + 1] * s;
    agg4[v * 4 + 2] = x[v * 4 + 2] * s;
    agg4[v * 4 + 3] = x[v * 4 + 3] * s;
}

__global__ void k_edge_agg4(const long long* __restrict__ ei,
                            const float* __restrict__ x,
                            const float* __restrict__ dinv,
                            float* __restrict__ agg4, int E) {
    int e = blockIdx.x * blockDim.x + threadIdx.x;
    if (e >= E) return;
    int s = (int)ei[e];
    int d = (int)ei[E + e];
    float norm = dinv[s] * dinv[d];
    const float4 xs = *(const float4*)(x + (size_t)s * 4);
    atomicAdd(&agg4[d * 4 + 0], xs.x * norm);
    atomicAdd(&agg4[d * 4 + 1], xs.y * norm);
    atomicAdd(&agg4[d * 4 + 2], xs.z * norm);
    atomicAdd(&agg4[d * 4 + 3], xs.w * norm);
}

__global__ void k_edge_agg2(const long long* __restrict__ ei,
                            const float* __restrict__ z,
                            const float* __restrict__ dinv,
                            float* __restrict__ agg2, int E) {
    int e = blockIdx.x * blockDim.x + threadIdx.x;
    if (e >= E) return;
    int s = (int)ei[e];
    int d = (int)ei[E + e];
    float norm = dinv[s] * dinv[d];
    atomicAdd(&agg2[d * 2 + 0], z[s * 2 + 0] * norm);
    atomicAdd(&agg2[d * 2 + 1], z[s * 2 + 1] * norm);
}

__global__ void k_logsoftmax(const float* __restrict__ agg2,
                             const float* __restrict__ b2,
                             float* __restrict__ out, int n) {
    int v = blockIdx.x * blockDim.x + threadIdx.x;
    if (v >= n) return;
    float a0 = agg2[v * 2 + 0] + b2[0];
    float a1 = agg2[v * 2 + 1] + b2[1];
    float m  = fmaxf(a0, a1);
    float lse = m + logf(__expf(a0 - m) + __expf(a1 - m));
    out[v * 2 + 0] = a0 - lse;
    out[v * 2 + 1] = a1 - lse;
}

// ---------------- fused WMMA layer-1 GEMM + ReLU + layer-2 projection ------
// One wave handles 16 nodes. h(16x128) = relu(agg4(16x4) @ W1(4x128) + b1)
// computed as 8 V_WMMA_F32_16X16X4_F32 tiles held in VGPRs, immediately
// contracted against W2(128x2) to z(16x2). Also writes the layer-2 self-loop
// init agg2[v] = z[v]*dinv[v]^2.
//
// VGPR layout assumptions (CDNA5 ISA 7.12.2, 32-bit matrices):
//   A 16x4 : lanes 0-15 row M=lane  {VGPR0=K0, VGPR1=K1}; lanes 16-31 {K2,K3}
//   B 4x16 : lanes 0-15 col N=lane  {VGPR0=K0, VGPR1=K1}; lanes 16-31 {K2,K3}
//   D 16x16: VGPR j = row j (lanes 0-15) / row j+8 (lanes 16-31), col=lane%16
__global__ __launch_bounds__(128) void k_l1_wmma_fused(
    const float* __restrict__ agg4, const float* __restrict__ W1,
    const float* __restrict__ b1,   const float* __restrict__ W2,
    const float* __restrict__ dinv, float* __restrict__ z,
    float* __restrict__ agg2, int n) {

    const int lane = threadIdx.x & 31;
    const int wave = (blockIdx.x * blockDim.x + threadIdx.x) >> 5;
    const int base = wave * 16;
    if (base >= n) return;                 // uniform per wave: EXEC stays all-1

    const int half = lane >> 4;            // 0: lanes 0-15, 1: lanes 16-31
    const int l16  = lane & 15;

    // A tile: row = base+l16 (clamped for the tail; stores are masked later)
    int arow = base + l16;
    arow = (arow < n) ? arow : (n - 1);
    v2f A;
    A.x = agg4[(size_t)arow * 4 + half * 2 + 0];
    A.y = agg4[(size_t)arow * 4 + half * 2 + 1];

    // Per-lane caches: this lane's 8 columns are col = t*16 + l16
    float bias[8], w2c0[8], w2c1[8];
#pragma unroll
    for (int t = 0; t < 8; ++t) {
        int col   = t * 16 + l16;
        bias[t]   = b1[col];
        w2c0[t]   = W2[col * 2 + 0];
        w2c1[t]   = W2[col * 2 + 1];
    }

    // Partial z for this lane's row group (rows j + 8*half), classes 0/1
    float p0[8], p1[8];
#pragma unroll
    for (int j = 0; j < 8; ++j) { p0[j] = 0.0f; p1[j] = 0.0f; }

#pragma unroll
    for (int t = 0; t < 8; ++t) {
        int col = t * 16 + l16;
        v2f B;
        B.x = W1[(half * 2 + 0) * 128 + col];
        B.y = W1[(half * 2 + 1) * 128 + col];
        v8f C = {};
        v8f D = __builtin_amdgcn_wmma_f32_16x16x4_f32(
            /*neg_a=*/false, A, /*neg_b=*/false, B,
            /*c_mod=*/(short)0, C, /*reuse_a=*/false, /*reuse_b=*/false);
#pragma unroll
        for (int j = 0; j < 8; ++j) {
            float h = fmaxf(D[j] + bias[t], 0.0f);   // bias + ReLU
            p0[j] = fmaf(h, w2c0[t], p0[j]);         // contract vs W2[:,0]
            p1[j] = fmaf(h, w2c1[t], p1[j]);         // contract vs W2[:,1]
        }
    }

    // Butterfly-reduce over the 16 lanes of each half (offsets < 16 keep
    // lanes within their half on wave32).
#pragma unroll
    for (int off = 1; off < 16; off <<= 1) {
#pragma unroll
        for (int j = 0; j < 8; ++j) {
            p0[j] += __shfl_xor(p0[j], off, 32);
            p1[j] += __shfl_xor(p1[j], off, 32);
        }
    }

    // Lanes with l16 < 8 own row = base + half*8 + l16 ; select p[l16]
    if (l16 < 8) {
        float z0 = 0.0f, z1 = 0.0f;
#pragma unroll
        for (int j = 0; j < 8; ++j) {
            if (l16 == j) { z0 = p0[j]; z1 = p1[j]; }
        }
        int row = base + half * 8 + l16;
        if (row < n) {
            float di = dinv[row];
            float s  = di * di;
            z[row * 2 + 0]    = z0;
            z[row * 2 + 1]    = z1;
            agg2[row * 2 + 0] = z0 * s;   // layer-2 self-loop init
            agg2[row * 2 + 1] = z1 * s;
        }
    }
}

// ---------------------------------------------------------------------------

extern "C" void kernel_launch(void* const* d_in, const int* in_sizes, int n_in,
                              void* d_out, int out_size, void* d_ws, size_t ws_size,
                              hipStream_t stream) {
    const float*     x  = (const float*)d_in[0];
    const long long* ei = (const long long*)d_in[1];   // int64 [2, E]
    const float*     W1 = (const float*)d_in[2];       // [4,128] row-major
    const float*     b1 = (const float*)d_in[3];       // [128]
    const float*     W2 = (const float*)d_in[4];       // [128,2] row-major
    const float*     b2 = (const float*)d_in[5];       // [2]
    float*           out = (float*)d_out;              // [N,2]

    const int N = in_sizes[0] / 4;
    const int E = in_sizes[1] / 2;

    // Workspace layout (floats): dinv[N] | agg4[4N] | z[2N] | agg2[2N]
    float* dinv = (float*)d_ws;
    float* agg4 = dinv + N;
    float* zbuf = agg4 + (size_t)4 * N;
    float* agg2 = zbuf + (size_t)2 * N;

    const int T = 256;
    const int gN = (N + T - 1) / T;
    const int gE = (E + T - 1) / T;

    // 1) deg = 0
    k_zero<<<gN, T, 0, stream>>>(dinv, N);
    // 2) deg via edge atomics
    k_degree<<<gE, T, 0, stream>>>(ei, dinv, E);
    // 3) dinv = rsqrt(deg+1) in place; agg4 init with self-loop x*dinv^2
    k_dinv_selfloop<<<gN, T, 0, stream>>>(dinv, x, agg4, N);
    // 4) 4-dim edge aggregation
    k_edge_agg4<<<gE, T, 0, stream>>>(ei, x, dinv, agg4, E);
    // 5) fused WMMA: h=relu(agg4@W1+b1) in regs, z=h@W2; agg2 self-loop init
    {
        int tiles = (N + 15) / 16;          // one wave per 16-node tile
        int waves_per_block = 128 / 32;
        int blocks = (tiles + waves_per_block - 1) / waves_per_block;
        k_l1_wmma_fused<<<blocks, 128, 0, stream>>>(agg4, W1, b1, W2, dinv,
                                                    zbuf, agg2, N);
    }
    // 6) 2-dim edge aggregation
    k_edge_agg2<<<gE, T, 0, stream>>>(ei, zbuf, dinv, agg2, E);
    // 7) bias + log_softmax over 2 classes
    k_logsoftmax<<<gN, T, 0, stream>>>(agg2, b2, out, N);
}